// MultiHeadAttention_5231270167139
// MI455X (gfx1250) — compile-verified
//
#include <hip/hip_runtime.h>
#include <hip/hip_bf16.h>
#include <math.h>

// Problem constants (from reference)
#define BB 4
#define SS 2048
#define DD 512
#define HH 8
#define DKK 64
#define DVV 64

typedef float v2f __attribute__((ext_vector_type(2)));
typedef float v8f __attribute__((ext_vector_type(8)));

// D = A(16x4) * B(4x16) + C, exact fp32 WMMA
__device__ __forceinline__ v8f wmma_f32(v2f a, v2f b, v8f c) {
  return __builtin_amdgcn_wmma_f32_16x16x4_f32(false, a, false, b, (short)0, c,
                                               false, false);
}

// ---------------------------------------------------------------------------
// Tensor Data Mover staging: DMA a (tile_h x tile_w) f32 tile with row stride
// `row_stride` (elements) from global into LDS (contiguous rows). One wave
// issues TENSOR_LOAD_TO_LDS and waits TENSORcnt; the barrier publishes it.
// Descriptor per CDNA5 ISA 8.3/8.4 (2-D, no padding, no iterate, no gather).
// This toolchain exposes the 6-arg builtin:
//   (uint32x4 g0, int32x8 g1, int32x4 g2, int32x4 g3, int32x8 g4, i32 cpol)
// ---------------------------------------------------------------------------
#if __has_builtin(__builtin_amdgcn_tensor_load_to_lds)
#define USE_TDM 1
#else
#define USE_TDM 0
#endif

#if USE_TDM
typedef unsigned int v4u __attribute__((ext_vector_type(4)));
typedef int v8i __attribute__((ext_vector_type(8)));
typedef int v4i __attribute__((ext_vector_type(4)));

__device__ __forceinline__ void tdm_load_2d(const float* gsrc, float* lds_dst,
                                            int tile_w, int tile_h,
                                            int row_stride) {
  const unsigned long long ga = (unsigned long long)gsrc;
  const unsigned int lds = (unsigned int)(unsigned long long)lds_dst;
  v4u g0;
  g0.x = 1u;                                         // count=1, user D#
  g0.y = lds;                                        // lds_addr (bytes)
  g0.z = (unsigned int)ga;                           // global_addr[31:0]
  g0.w = ((unsigned int)(ga >> 32) & 0x01FFFFFFu) |  // global_addr[56:32]
         (2u << 30);                                 // type=2 ("image")
  v8i g1;
  g1[0] = (2 << 16);                                 // data_size=4B; mask=0
  g1[1] = (tile_w & 0xFFFF) << 16;                   // tensor_dim0[15:0]
  g1[2] = ((unsigned)tile_w >> 16) |                 // tensor_dim0[31:16]
          ((tile_h & 0xFFFF) << 16);                 // tensor_dim1[15:0]
  g1[3] = (((unsigned)tile_h >> 16) & 0xFFFF) |      // tensor_dim1[31:16]
          ((tile_w & 0xFFFF) << 16);                 // tile_dim0
  g1[4] = (tile_h & 0xFFFF);                         // tile_dim1; tile_dim2=0
  g1[5] = row_stride;                                // tensor_dim0_stride lo32
  g1[6] = 0;                                         // stride hi; dim1_stride
  g1[7] = 0;
  v4i z4 = {0, 0, 0, 0};
  v8i z8 = {0, 0, 0, 0, 0, 0, 0, 0};
  __builtin_amdgcn_tensor_load_to_lds(g0, g1, z4, z4, z8, 0);
}
#endif

template <int TW>
__device__ __forceinline__ void stage_tile_16rows(const float* g, float* lds,
                                                  int row_stride, int tid,
                                                  int wave) {
#if USE_TDM
  if (wave == 0) {
    tdm_load_2d(g, lds, TW, 16, row_stride);
    __builtin_amdgcn_s_wait_tensorcnt(0);
  }
#else
  for (int i = tid; i < 16 * TW; i += 256) {
    const int r = i / TW, c = i % TW;
    lds[i] = g[(size_t)r * row_stride + c];
  }
#endif
  __syncthreads();
}

// ---------------------------------------------------------------------------
// Kernel 1: C[M,N] = A[M,K] @ W[K,N], M=8192, K=N=512.
// grid = (M/16, N/128), block = 256 (8 waves). Wave w owns 16x16 tile at
// n0 = blockIdx.y*128 + w*16. A block (16 x 512) staged in LDS via TDM.
// ---------------------------------------------------------------------------
__global__ void __launch_bounds__(256) proj_gemm(const float* __restrict__ A,
                                                 const float* __restrict__ W,
                                                 float* __restrict__ C) {
  const int K = DD, N = DD;
  __shared__ float sA[16 * DD];  // 32 KB
  const int tid = threadIdx.x;
  const int wave = tid >> 5;
  const int lane = tid & 31;
  const int m0 = blockIdx.x * 16;
  const int n0 = blockIdx.y * 128 + wave * 16;

  stage_tile_16rows<DD>(A + (size_t)m0 * K, sA, K, tid, wave);

  const int nr = lane & 15;   // N col (B/C) and M row (A)
  const int kh = lane >> 4;   // K sub-pair selector
  v8f acc = {};
  for (int k = 0; k < K; k += 4) {
    v2f a, b;
    a.x = sA[nr * K + k + 2 * kh];
    a.y = sA[nr * K + k + 2 * kh + 1];
    b.x = W[(size_t)(k + 2 * kh) * N + n0 + nr];
    b.y = W[(size_t)(k + 2 * kh + 1) * N + n0 + nr];
    acc = wmma_f32(a, b, acc);
  }
#pragma unroll
  for (int j = 0; j < 8; ++j)
    C[(size_t)(m0 + j + 8 * kh) * N + (n0 + nr)] = acc[j];
}

// ---------------------------------------------------------------------------
// Kernel 2: fused scores -> mask -> softmax -> attn store -> attn @ V.
// grid = (S/16, H, B), block = 256. 16x2048 f32 score strip lives in LDS
// (128 KB; CDNA5 WGP has 320 KB). attn is written to HBM exactly once.
// ---------------------------------------------------------------------------
__global__ void __launch_bounds__(256) attn_fused(
    const float* __restrict__ Qp, const float* __restrict__ Kp,
    const float* __restrict__ Vp, const unsigned char* __restrict__ mask,
    float* __restrict__ attn_out, float* __restrict__ prob_out) {
  __shared__ float sS[16 * SS];   // 128 KB score strip
  __shared__ float sQ[16 * DKK];  // 4 KB Q block; reused as split-K buffer

  const int tid = threadIdx.x;
  const int wave = tid >> 5;
  const int lane = tid & 31;
  const int q0 = blockIdx.x * 16;
  const int h = blockIdx.y;
  const int b = blockIdx.z;
  const int stride = HH * DKK;  // 512, row stride of projected [B,S,H*64]

  const float* Qb = Qp + ((size_t)b * SS + q0) * stride + h * DKK;
  const float* Kb = Kp + (size_t)b * SS * stride + h * DKK;
  const float* Vb = Vp + (size_t)b * SS * stride + h * DVV;

  stage_tile_16rows<DKK>(Qb, sQ, stride, tid, wave);

  const int nr = lane & 15;
  const int kh = lane >> 4;

  // ---- scores = Q @ K^T * 1/8 : 128 column tiles split across 8 waves ----
  for (int t = wave; t < SS / 16; t += 8) {
    const int s0 = t * 16;
    v8f acc = {};
    const float* kr = Kb + (size_t)(s0 + nr) * stride;  // K row for col nr
    for (int k = 0; k < DKK; k += 4) {
      v2f a, bb;
      a.x = sQ[nr * DKK + k + 2 * kh];
      a.y = sQ[nr * DKK + k + 2 * kh + 1];
      bb.x = kr[k + 2 * kh];      // B[d][s] = K[s][d]
      bb.y = kr[k + 2 * kh + 1];
      acc = wmma_f32(a, bb, acc);
    }
#pragma unroll
    for (int j = 0; j < 8; ++j)
      sS[(j + 8 * kh) * SS + s0 + nr] = acc[j] * 0.125f;
  }
  __syncthreads();

  // ---- mask + softmax: one row per 16-lane half-wave (wave32) ----
  {
    const int row = tid >> 4;  // 0..15
    const int cl = tid & 15;
    float* srow = sS + row * SS;
    const unsigned char* mrow = mask + ((size_t)b * SS + q0 + row) * SS;
    float mx = -INFINITY;
    for (int c = cl; c < SS; c += 16) {
      float v = srow[c];
      if (mrow[c]) v = -1e9f;
      srow[c] = v;
      mx = fmaxf(mx, v);
    }
    for (int o = 8; o; o >>= 1) mx = fmaxf(mx, __shfl_xor(mx, o));
    float sum = 0.f;
    for (int c = cl; c < SS; c += 16) {
      float e = expf(srow[c] - mx);
      srow[c] = e;
      sum += e;
    }
    for (int o = 8; o; o >>= 1) sum += __shfl_xor(sum, o);
    const float inv = 1.0f / sum;
    float* arow = attn_out + (((size_t)b * HH + h) * SS + q0 + row) * SS;
    for (int c = cl; c < SS; c += 16) {
      float p = srow[c] * inv;
      srow[c] = p;
      arow[c] = p;  // the single HBM write of attn
    }
  }
  __syncthreads();

  // ---- prob = attn(16x2048) @ V(2048x64): 4 col tiles x split-K halves ----
  const int vt = wave & 3;     // 16-col tile of V
  const int khalf = wave >> 2; // K half (0: 0..1023, 1: 1024..2047)
  v8f acc = {};
  for (int k = khalf * (SS / 2); k < (khalf + 1) * (SS / 2); k += 4) {
    v2f a, bb;
    a.x = sS[nr * SS + k + 2 * kh];
    a.y = sS[nr * SS + k + 2 * kh + 1];
    const float* vr = Vb + (size_t)(k + 2 * kh) * stride + vt * 16 + nr;
    bb.x = vr[0];
    bb.y = vr[stride];
    acc = wmma_f32(a, bb, acc);
  }
  __syncthreads();  // sQ no longer needed as Q; reuse for split-K reduction
  if (khalf == 0) {
#pragma unroll
    for (int j = 0; j < 8; ++j)
      sQ[(j + 8 * kh) * DVV + vt * 16 + nr] = acc[j];
  }
  __syncthreads();
  if (khalf == 1) {
#pragma unroll
    for (int j = 0; j < 8; ++j) {
      float r = sQ[(j + 8 * kh) * DVV + vt * 16 + nr] + acc[j];
      // store heads concatenated: [B, S, H*DV] so W_O GEMM reads row-major
      prob_out[((size_t)b * SS + q0 + j + 8 * kh) * (HH * DVV) + h * DVV +
               vt * 16 + nr] = r;
    }
  }
}

// ---------------------------------------------------------------------------
// Kernel 3: out = LN(prob @ W_O + input_Q) * gamma + beta.
// grid = B*S/16 = 512, block 256. 16x512 result block in LDS, then per-row LN.
// ---------------------------------------------------------------------------
__global__ void __launch_bounds__(256) out_ln(const float* __restrict__ P,
                                              const float* __restrict__ WO,
                                              const float* __restrict__ X,
                                              const float* __restrict__ gamma,
                                              const float* __restrict__ beta,
                                              float* __restrict__ out) {
  __shared__ float sA[16 * DD];  // prob block, 32 KB
  __shared__ float sY[16 * DD];  // result block, 32 KB
  const int tid = threadIdx.x;
  const int wave = tid >> 5;
  const int lane = tid & 31;
  const int m0 = blockIdx.x * 16;

  stage_tile_16rows<DD>(P + (size_t)m0 * DD, sA, DD, tid, wave);

  const int nr = lane & 15;
  const int kh = lane >> 4;
  for (int t = wave; t < DD / 16; t += 8) {
    const int n0 = t * 16;
    v8f acc = {};
    for (int k = 0; k < DD; k += 4) {
      v2f a, b;
      a.x = sA[nr * DD + k + 2 * kh];
      a.y = sA[nr * DD + k + 2 * kh + 1];
      b.x = WO[(size_t)(k + 2 * kh) * DD + n0 + nr];
      b.y = WO[(size_t)(k + 2 * kh + 1) * DD + n0 + nr];
      acc = wmma_f32(a, b, acc);
    }
#pragma unroll
    for (int j = 0; j < 8; ++j) {
      const int r = j + 8 * kh;
      sY[r * DD + n0 + nr] =
          acc[j] + X[(size_t)(m0 + r) * DD + n0 + nr];  // residual
    }
  }
  __syncthreads();

  // LayerNorm: one row per 16-lane half-wave
  const int row = tid >> 4;
  const int cl = tid & 15;
  const float* yrow = sY + row * DD;
  float s = 0.f;
  for (int c = cl; c < DD; c += 16) s += yrow[c];
  for (int o = 8; o; o >>= 1) s += __shfl_xor(s, o);
  const float mean = s * (1.0f / DD);
  float v = 0.f;
  for (int c = cl; c < DD; c += 16) {
    float d = yrow[c] - mean;
    v += d * d;
  }
  for (int o = 8; o; o >>= 1) v += __shfl_xor(v, o);
  const float rstd = rsqrtf(v * (1.0f / DD) + 1e-5f);
  float* orow = out + (size_t)(m0 + row) * DD;
  for (int c = cl; c < DD; c += 16)
    orow[c] = (yrow[c] - mean) * rstd * gamma[c] + beta[c];
}

// ---------------------------------------------------------------------------
extern "C" void kernel_launch(void* const* d_in, const int* in_sizes, int n_in,
                              void* d_out, int out_size, void* d_ws,
                              size_t ws_size, hipStream_t stream) {
  const float* inQ = (const float*)d_in[0];
  const float* inK = (const float*)d_in[1];
  const float* inV = (const float*)d_in[2];
  const unsigned char* mask = (const unsigned char*)d_in[3];
  const float* WQ = (const float*)d_in[4];
  const float* WK = (const float*)d_in[5];
  const float* WV = (const float*)d_in[6];
  const float* WO = (const float*)d_in[7];
  const float* gamma = (const float*)d_in[8];
  const float* beta = (const float*)d_in[9];

  float* out = (float*)d_out;                        // [B,S,D]
  float* attn = out + (size_t)BB * SS * DD;          // [B,H,S,S]

  const size_t npr = (size_t)BB * SS * DD;           // 4,194,304 floats
  float* wsQ = (float*)d_ws;
  float* wsK = wsQ + npr;
  float* wsV = wsK + npr;
  float* wsP = wsV + npr;                            // total 64 MB

  dim3 blk(256);
  dim3 gproj(BB * SS / 16, DD / 128);                // (512, 4)
  hipLaunchKernelGGL(proj_gemm, gproj, blk, 0, stream, inQ, WQ, wsQ);
  hipLaunchKernelGGL(proj_gemm, gproj, blk, 0, stream, inK, WK, wsK);
  hipLaunchKernelGGL(proj_gemm, gproj, blk, 0, stream, inV, WV, wsV);

  dim3 gattn(SS / 16, HH, BB);                       // (128, 8, 4)
  hipLaunchKernelGGL(attn_fused, gattn, blk, 0, stream, wsQ, wsK, wsV, mask,
                     attn, wsP);

  dim3 gout(BB * SS / 16);                           // (512)
  hipLaunchKernelGGL(out_ln, gout, blk, 0, stream, wsP, WO, inQ, gamma, beta,
                     out);
}